// CanonicalMnist_29180007809183
// MI455X (gfx1250) — compile-verified
//
#include <hip/hip_runtime.h>

// ---------------------------------------------------------------------------
// CanonicalMnist on MI455X (gfx1250)
//   GEMM1+relu+GEMM2 fused, fp32 WMMA (V_WMMA_F32_16X16X4_F32).
//   Staging via Tensor Data Mover (TENSOR_LOAD_TO_LDS) with LDS padding so
//   every WMMA fragment is one aligned 8-byte conflict-light ds_load.
//   ~208 MB HBM traffic @ 23.3 TB/s => ~9us floor; fp32 matrix pipe keeps
//   reference precision and still lands near that floor (AI ~ 64 FLOP/B).
// ---------------------------------------------------------------------------

typedef float v2f __attribute__((ext_vector_type(2)));
typedef float v8f __attribute__((ext_vector_type(8)));
typedef unsigned int u32x4 __attribute__((ext_vector_type(4)));
typedef int i32x8 __attribute__((ext_vector_type(8)));
typedef int i32x4 __attribute__((ext_vector_type(4)));

#define D_IN   784
#define D_HID  128
#define D_OUT  10
#define NCAN   16
#define KC     112            // K-chunk (784 = 7*112; 112 = 7*16 -> TDM pad stays row-aligned)
#define NCHUNK 7
#define ROWS_PER_BLOCK 256    // 8 waves x 2 M-tiles x 16 rows
#define PADDED_ROW 126        // 112 data dwords + 7 intervals * 2 pad dwords
#define XS_DW   (ROWS_PER_BLOCK * PADDED_ROW)   // 32256 dwords
#define W1S_DW  (D_HID * PADDED_ROW)            // 16128 dwords
#define HS_STRIDE 129

// workspace layout (floats)
#define WS_W1 0                        // [128][784] combined fc weight
#define WS_B1 (WS_W1 + D_HID * D_IN)   // [128]
#define WS_W2 (WS_B1 + D_HID)          // [16][128] combined cls weight, rows 10..15 = 0
#define WS_B2 (WS_W2 + 16 * D_HID)     // [16], 10..15 = 0
#define WS_TOTAL (WS_B2 + 16)

// ---------------------------------------------------------------------------
// Kernel A: fold ensemble factor into single weight set (tiny)
// ---------------------------------------------------------------------------
__global__ void combine_weights_kernel(const float* __restrict__ fc_w,
                                       const float* __restrict__ fc_b,
                                       const float* __restrict__ cls_w,
                                       const float* __restrict__ cls_b,
                                       const float* __restrict__ factor,
                                       float* __restrict__ ws) {
  const int tid = blockIdx.x * blockDim.x + threadIdx.x;
  const int n_w1 = D_HID * D_IN;        // 100352
  const int n_b1 = D_HID;
  const int n_w2 = 16 * D_HID;          // padded
  if (tid < n_w1) {
    float s = 0.f;
#pragma unroll
    for (int k = 0; k < NCAN; ++k) s += factor[k] * fc_w[k * n_w1 + tid];
    ws[WS_W1 + tid] = s;
  } else if (tid < n_w1 + n_b1) {
    const int o = tid - n_w1;
    float s = 0.f;
#pragma unroll
    for (int k = 0; k < NCAN; ++k) s += factor[k] * fc_b[k * D_HID + o];
    ws[WS_B1 + o] = s;
  } else if (tid < n_w1 + n_b1 + n_w2) {
    const int e = tid - (n_w1 + n_b1);
    const int o = e / D_HID, d = e % D_HID;
    float s = 0.f;
    if (o < D_OUT) {
#pragma unroll
      for (int k = 0; k < NCAN; ++k) s += factor[k] * cls_w[(k * D_OUT + o) * D_HID + d];
    }
    ws[WS_W2 + e] = s;
  } else if (tid < n_w1 + n_b1 + n_w2 + 16) {
    const int o = tid - (n_w1 + n_b1 + n_w2);
    float s = 0.f;
    if (o < D_OUT) {
#pragma unroll
      for (int k = 0; k < NCAN; ++k) s += factor[k] * cls_b[k * D_OUT + o];
    }
    ws[WS_B2 + o] = s;
  }
}

// ---------------------------------------------------------------------------
// TDM 2D tile load descriptor (D# groups per CDNA5 ISA ch.8), data_size = 4B,
// padding: 2 dwords after every 16 data dwords -> padded row = 126 dwords.
// 6-arg clang-23 builtin form:
//   (u32x4 g0, i32x8 g1, i32x4 g2, i32x4 g3, i32x8 extra, i32 cpol)
// ---------------------------------------------------------------------------
__device__ __forceinline__ void tdm_load_2d(unsigned lds_byte_addr, const void* gaddr,
                                            unsigned tensor_d0, unsigned tensor_d1,
                                            unsigned tile_d0, unsigned tile_d1,
                                            unsigned d0_stride) {
  const unsigned long long ga = (unsigned long long)(uintptr_t)gaddr;
  u32x4 g0;
  g0[0] = 1u;                                            // count=1, user descriptor
  g0[1] = lds_byte_addr;                                 // lds_addr [63:32]
  g0[2] = (unsigned)(ga & 0xFFFFFFFFu);                  // global_addr [95:64]
  g0[3] = (unsigned)((ga >> 32) & 0x01FFFFFFu)           // global_addr [120:96]
          | (2u << 30);                                  // type=2 ("image") [127:126]
  i32x8 g1;
  g1[0] = (2 << 16)            // data_size = 4 bytes
        | (1 << 20)            // pad_enable
        | (3 << 22)            // pad_interval code 3 = 16 dwords
        | (1 << 25);           // pad_amount  code 1 = 2 dwords
  g1[1] = (int)((tensor_d0 & 0xFFFFu) << 16);                                  // dim0[15:0]
  g1[2] = (int)(((tensor_d0 >> 16) & 0xFFFFu) | ((tensor_d1 & 0xFFFFu) << 16));// dim0[31:16], dim1[15:0]
  g1[3] = (int)(((tensor_d1 >> 16) & 0xFFFFu) | ((tile_d0 & 0xFFFFu) << 16));  // dim1[31:16], tile0
  g1[4] = (int)(tile_d1 & 0xFFFFu);                                            // tile1 (tile2=0)
  g1[5] = (int)d0_stride;                                                      // dim0_stride[31:0]
  g1[6] = 0;
  g1[7] = 0;
  const i32x4 gz4 = {0, 0, 0, 0};
  const i32x8 gz8 = {0, 0, 0, 0, 0, 0, 0, 0};
  __builtin_amdgcn_tensor_load_to_lds(g0, g1, gz4, gz4, gz8, 0);
}

// ---------------------------------------------------------------------------
// Kernel B: fused MLP. 256 threads = 8 wave32; each wave: 2 M-tiles (32 rows)
// x 8 N-tiles; 16 fp32 WMMA accumulators held in 128 VGPRs.
// Dynamic LDS 193536 B:
//   phase1: xs[256 rows][126 dw] + w1s[128 rows][126 dw]   (TDM-filled)
//   phase2: hs[256][129] + w2s[128][16]                    (reuses same memory)
// ---------------------------------------------------------------------------
__global__ __launch_bounds__(256) void mlp_wmma_kernel(const float* __restrict__ x,
                                                       const float* __restrict__ ws,
                                                       float* __restrict__ out) {
  extern __shared__ float lds[];
  float* xs  = lds;            // XS_DW dwords
  float* w1s = lds + XS_DW;    // W1S_DW dwords
  float* hs  = lds;            // 256*129 dwords (phase 2)
  float* w2s = lds + 256 * HS_STRIDE;  // 128*16 dwords (phase 2)

  const int tid  = threadIdx.x;
  const int lane = tid & 31;
  const int wv   = tid >> 5;     // wave id 0..7
  const int l16  = lane & 15;
  const int sel  = lane >> 4;    // half-wave select
  const int row0 = blockIdx.x * ROWS_PER_BLOCK;

  const float* w1c = ws + WS_W1;
  const float* b1c = ws + WS_B1;
  const float* w2c = ws + WS_W2;
  const float* b2c = ws + WS_B2;

  float b1v[8];
#pragma unroll
  for (int nt = 0; nt < 8; ++nt) b1v[nt] = b1c[nt * 16 + l16];

  const v8f vzero = {0.f, 0.f, 0.f, 0.f, 0.f, 0.f, 0.f, 0.f};
  v8f accA[8], accB[8];
#pragma unroll
  for (int nt = 0; nt < 8; ++nt) { accA[nt] = vzero; accB[nt] = vzero; }

  const unsigned xs_lds  = (unsigned)(uintptr_t)xs;   // generic->LDS: low 32 bits
  const unsigned w1s_lds = (unsigned)(uintptr_t)w1s;

  // per-lane fragment row bases (dword offsets into padded tiles)
  const float* xr0 = xs + (wv * 32 + l16) * PADDED_ROW;        // A tile 0 row
  const float* xr1 = xs + (wv * 32 + 16 + l16) * PADDED_ROW;   // A tile 1 row
  const float* wr  = w1s + l16 * PADDED_ROW;                   // B row base (+nt*16 rows)

  for (int kc = 0; kc < NCHUNK; ++kc) {
    __syncthreads();   // previous chunk's LDS reads complete
    if (wv == 0) {
      // x chunk: 256 rows x 112 cols from [65536][784]
      tdm_load_2d(xs_lds, x + (size_t)row0 * D_IN + kc * KC,
                  D_IN, 65536u, KC, ROWS_PER_BLOCK, D_IN);
    } else if (wv == 1) {
      // W1 chunk: 128 rows x 112 cols from [128][784]
      tdm_load_2d(w1s_lds, w1c + kc * KC,
                  D_IN, D_HID, KC, D_HID, D_IN);
    }
    __builtin_amdgcn_s_wait_tensorcnt(0);   // no-op for non-issuing waves
    __syncthreads();

    for (int k4 = 0; k4 < KC; k4 += 4) {
      const int kk   = k4 + 2 * sel;
      const int coff = (kk >> 4) * 18 + (kk & 15);   // padded column offset (8B aligned)
      // A fragments (16x4 f32): lane holds (M=l16, K=kk..kk+1)
      const v2f a0 = *(const v2f*)(xr0 + coff);
      const v2f a1 = *(const v2f*)(xr1 + coff);
      // Batch all 8 B fragments into distinct registers so the 8 ds_load_b64
      // issue back-to-back and s_wait_dscnt can drain progressively instead of
      // a full wait before every WMMA pair.
      v2f bf[8];
#pragma unroll
      for (int nt = 0; nt < 8; ++nt) {
        bf[nt] = *(const v2f*)(wr + nt * 16 * PADDED_ROW + coff);
      }
#pragma unroll
      for (int nt = 0; nt < 8; ++nt) {
        accA[nt] = __builtin_amdgcn_wmma_f32_16x16x4_f32(
            false, a0, false, bf[nt], (short)0, accA[nt], false, false);
        accB[nt] = __builtin_amdgcn_wmma_f32_16x16x4_f32(
            false, a1, false, bf[nt], (short)0, accB[nt], false, false);
      }
    }
  }

  // ---- epilogue: bias + relu, spill h (32x128 per wave) to LDS; stage W2 ----
  __syncthreads();   // all phase-1 LDS reads done; safe to overwrite
  {
    float* hw0 = hs + (wv * 32) * HS_STRIDE + l16;
    float* hw1 = hs + (wv * 32 + 16) * HS_STRIDE + l16;
#pragma unroll
    for (int nt = 0; nt < 8; ++nt) {
#pragma unroll
      for (int i = 0; i < 8; ++i) {
        const int m = (i + 8 * sel) * HS_STRIDE + nt * 16;
        float ha = accA[nt][i] + b1v[nt];
        float hb = accB[nt][i] + b1v[nt];
        hw0[m] = ha > 0.f ? ha : 0.f;
        hw1[m] = hb > 0.f ? hb : 0.f;
      }
    }
    for (int e = tid; e < 16 * D_HID; e += 256) {
      const int o = e >> 7, d = e & 127;      // coalesced read of w2c[o][d]
      w2s[d * 16 + o] = w2c[e];               // w2s layout [K][16]
    }
  }
  __syncthreads();

  // ---- GEMM2: two 16x16 output tiles per wave, K = 128 ----
  v8f c0 = vzero, c1 = vzero;
  const float* hr0 = hs + (wv * 32 + l16) * HS_STRIDE;
  const float* hr1 = hr0 + 16 * HS_STRIDE;
  for (int k4 = 0; k4 < D_HID; k4 += 4) {
    const int kk = k4 + 2 * sel;
    v2f a0, a1, b;
    a0.x = hr0[kk];          a0.y = hr0[kk + 1];
    a1.x = hr1[kk];          a1.y = hr1[kk + 1];
    b.x  = w2s[kk * 16 + l16];
    b.y  = w2s[(kk + 1) * 16 + l16];
    c0 = __builtin_amdgcn_wmma_f32_16x16x4_f32(false, a0, false, b, (short)0, c0, false, false);
    c1 = __builtin_amdgcn_wmma_f32_16x16x4_f32(false, a1, false, b, (short)0, c1, false, false);
  }

  const float b2v = (l16 < D_OUT) ? b2c[l16] : 0.f;
  if (l16 < D_OUT) {
#pragma unroll
    for (int i = 0; i < 8; ++i) {
      const int m = i + 8 * sel;
      out[(size_t)(row0 + wv * 32 + m) * D_OUT + l16] = c0[i] + b2v;
      out[(size_t)(row0 + wv * 32 + 16 + m) * D_OUT + l16] = c1[i] + b2v;
    }
  }
}

// ---------------------------------------------------------------------------
extern "C" void kernel_launch(void* const* d_in, const int* in_sizes, int n_in,
                              void* d_out, int out_size, void* d_ws, size_t ws_size,
                              hipStream_t stream) {
  const float* x      = (const float*)d_in[0];
  const float* fc_w   = (const float*)d_in[1];
  const float* fc_b   = (const float*)d_in[2];
  const float* cls_w  = (const float*)d_in[3];
  const float* cls_b  = (const float*)d_in[4];
  const float* factor = (const float*)d_in[5];
  float* out = (float*)d_out;
  float* ws  = (float*)d_ws;

  combine_weights_kernel<<<(WS_TOTAL + 255) / 256, 256, 0, stream>>>(
      fc_w, fc_b, cls_w, cls_b, factor, ws);

  // phase1 LDS (48384 dwords) dominates phase2 (35072 dwords)
  const size_t lds_bytes = (size_t)(XS_DW + W1S_DW) * sizeof(float);  // 193536 B
  mlp_wmma_kernel<<<65536 / ROWS_PER_BLOCK, 256, lds_bytes, stream>>>(x, ws, out);
}